// TPTT_RNN_71176198029928
// MI455X (gfx1250) — compile-verified
//
#include <hip/hip_runtime.h>
#include <hip/hip_bf16.h>
#include <math.h>

typedef __attribute__((ext_vector_type(2))) float v2f;
typedef __attribute__((ext_vector_type(8))) float v8f;

#define S_LEN 2048
#define BATCH 128
#define N_INP 64
#define N_HID 128
#define N_OUT 10

// -------- Phase 1: xW[s,b,:] = x[s,b,:] @ Wxh + bh  (written into h-output region) --------
// Grid: (S*B)/16 blocks, 256 threads (8 waves). Block handles a 16-row M tile;
// wave w computes the 16x16 output tile at N = 16*w via 16 chained f32 WMMAs (K=64).
__global__ void __launch_bounds__(256)
rnn_xproj_kernel(const float* __restrict__ x, const float* __restrict__ Wxh,
                 const float* __restrict__ bh, float* __restrict__ xw_out) {
    __shared__ float xs[16 * 68];  // padded stride 68 -> conflict-free column reads
    const int tid  = threadIdx.x;
    const int row0 = blockIdx.x * 16;  // row in flattened [S*B]

    // cooperative, coalesced load of the 16x64 x tile
    for (int i = tid; i < 16 * N_INP; i += 256) {
        int r = i >> 6, c = i & 63;
        xs[r * 68 + c] = x[(size_t)(row0 + r) * N_INP + c];
    }
    __syncthreads();

    const int wave = tid >> 5;
    const int lane = tid & 31;
    const int m  = lane & 15;   // A-matrix row / B,C column-within-tile
    const int kh = lane >> 4;   // K half select
    const int n  = wave * 16 + m;

    v8f c;
    const float bias = bh[n];
#pragma unroll
    for (int v = 0; v < 8; ++v) c[v] = bias;

#pragma unroll
    for (int kb = 0; kb < 16; ++kb) {
        const int k = kb * 4 + kh * 2;
        v2f a, b;
        a.x = xs[m * 68 + k];
        a.y = xs[m * 68 + k + 1];
        b.x = Wxh[(size_t)k       * N_HID + n];
        b.y = Wxh[(size_t)(k + 1) * N_HID + n];
        c = __builtin_amdgcn_wmma_f32_16x16x4_f32(false, a, false, b,
                                                  (short)0, c, false, false);
    }

#pragma unroll
    for (int v = 0; v < 8; ++v)
        xw_out[(size_t)(row0 + v + 8 * kh) * N_HID + n] = c[v];
}

// -------- Phase 2: sequential scan h_t = tanh(h_{t-1} @ Whh + xW_t) --------
// 8 blocks, each owns batch rows [16*bg, 16*bg+16) (independent recurrences).
// Whh (64KB) + a double-buffered h slice live in LDS for all 2048 steps. Wave w
// computes the N=16*w tile; K=128 accumulation split into two independent
// 16-deep WMMA chains. xW_t is read from the output slot and overwritten with
// h_t (in-place); the xW tile for step s+1 is prefetched into registers during
// step s's WMMA chain to take the global-load latency off the serial path.
// Double buffering cuts the per-step barrier count to one.
__global__ void __launch_bounds__(256)
rnn_scan_kernel(const float* __restrict__ Whh, float* __restrict__ h_out) {
    __shared__ float whs[N_HID * N_HID]; // 64 KB
    __shared__ float hs[2][16 * 132];    // padded stride 132 -> conflict-free A reads

    const int tid = threadIdx.x;
    const int b0  = blockIdx.x * 16;

    for (int i = tid; i < N_HID * N_HID; i += 256) whs[i] = Whh[i];
    for (int i = tid; i < 16 * 132; i += 256) hs[0][i] = 0.0f;
    __syncthreads();

    const int wave = tid >> 5;
    const int lane = tid & 31;
    const int m    = lane & 15;
    const int kh   = lane >> 4;
    const int n    = wave * 16 + m;
    const int crow = 8 * kh;   // C/D row base for this lane half

    // prefetch the xW tile for step 0
    float xw[8];
    {
        const float* base0 = h_out + (size_t)b0 * N_HID;
#pragma unroll
        for (int v = 0; v < 8; ++v)
            xw[v] = base0[(size_t)(v + crow) * N_HID + n];
    }

    for (int s = 0; s < S_LEN; ++s) {
        const int cur = s & 1, nxt = cur ^ 1;
        float* base = h_out + ((size_t)s * BATCH + b0) * N_HID;

        // c0 starts from the prefetched xW tile; c1 is the second chain
        v8f c0, c1 = {};
#pragma unroll
        for (int v = 0; v < 8; ++v) c0[v] = xw[v];

        // issue next step's xW loads now; they overlap the WMMA chain below
        if (s + 1 < S_LEN) {
            const float* nb = base + (size_t)BATCH * N_HID;
#pragma unroll
            for (int v = 0; v < 8; ++v)
                xw[v] = nb[(size_t)(v + crow) * N_HID + n];
        }

#pragma unroll
        for (int kb = 0; kb < 32; kb += 2) {
            const int k0 = kb * 4 + kh * 2;
            const int k1 = k0 + 4;
            v2f a0, b0v, a1, b1v;
            a0.x  = hs[cur][m * 132 + k0];
            a0.y  = hs[cur][m * 132 + k0 + 1];
            b0v.x = whs[k0       * N_HID + n];
            b0v.y = whs[(k0 + 1) * N_HID + n];
            a1.x  = hs[cur][m * 132 + k1];
            a1.y  = hs[cur][m * 132 + k1 + 1];
            b1v.x = whs[k1       * N_HID + n];
            b1v.y = whs[(k1 + 1) * N_HID + n];
            c0 = __builtin_amdgcn_wmma_f32_16x16x4_f32(false, a0, false, b0v,
                                                       (short)0, c0, false, false);
            c1 = __builtin_amdgcn_wmma_f32_16x16x4_f32(false, a1, false, b1v,
                                                       (short)0, c1, false, false);
        }

#pragma unroll
        for (int v = 0; v < 8; ++v) {
            const float val = tanhf(c0[v] + c1[v]);
            hs[nxt][(v + crow) * 132 + n] = val;              // h_t for next step
            base[(size_t)(v + crow) * N_HID + n] = val;       // h output (in-place)
        }

        // single barrier: all reads of hs[cur] and writes of hs[nxt] complete
        __syncthreads();
    }
}

// -------- Phase 3: logits = h_last @ Why + by  (tiny; plain VALU) --------
__global__ void rnn_head_kernel(const float* __restrict__ h_last,
                                const float* __restrict__ Why,
                                const float* __restrict__ by,
                                float* __restrict__ logits) {
    const int idx = blockIdx.x * blockDim.x + threadIdx.x;
    if (idx >= BATCH * N_OUT) return;
    const int b = idx / N_OUT, o = idx % N_OUT;
    float acc = by[o];
#pragma unroll 8
    for (int k = 0; k < N_HID; ++k)
        acc = fmaf(h_last[b * N_HID + k], Why[k * N_OUT + o], acc);
    logits[idx] = acc;
}

extern "C" void kernel_launch(void* const* d_in, const int* in_sizes, int n_in,
                              void* d_out, int out_size, void* d_ws, size_t ws_size,
                              hipStream_t stream) {
    const float* x   = (const float*)d_in[0];
    const float* Wxh = (const float*)d_in[1];
    const float* Whh = (const float*)d_in[2];
    const float* Why = (const float*)d_in[3];
    const float* bh  = (const float*)d_in[4];
    const float* by  = (const float*)d_in[5];

    float* logits = (float*)d_out;                 // [128, 10]
    float* h      = logits + BATCH * N_OUT;        // [2048, 128, 128]

    // Phase 1: xW -> h region (in-place staging)
    rnn_xproj_kernel<<<(S_LEN * BATCH) / 16, 256, 0, stream>>>(x, Wxh, bh, h);

    // Phase 2: sequential scan, consumes xW from h region, overwrites with h_t
    rnn_scan_kernel<<<BATCH / 16, 256, 0, stream>>>(Whh, h);

    // Phase 3: head on h_last
    const float* h_last = h + (size_t)(S_LEN - 1) * BATCH * N_HID;
    rnn_head_kernel<<<(BATCH * N_OUT + 255) / 256, 256, 0, stream>>>(h_last, Why, by, logits);
}